// MultiboxLoss_WideSamples_70325794505518
// MI455X (gfx1250) — compile-verified
//
#include <hip/hip_runtime.h>
#include <math.h>

// MultiboxLoss for MI455X (gfx1250, wave32).
// B=128 rows, one 1024-thread workgroup per row; single streaming pass over
// HBM (~139MB => ~6us roofline floor), top-K hard-negative mining done fully
// in LDS via 32-step radix select on monotone float keys. All cross-lane /
// cross-wave f32 reductions are implemented with V_WMMA_F32_16X16X4_F32
// (B = ones => matrix op computes 32-lane sums at full f32 precision).

#define BATCH     128
#define NPRIOR    8732
#define NCLS      21
#define NEG_RATIO 3
#define BLOCK     1024
#define NWAVES    (BLOCK / 32)

typedef __attribute__((ext_vector_type(2))) float v2f;
typedef __attribute__((ext_vector_type(8))) float v8f;

// Sum of one float per lane across the full wave32, returned to every lane.
// A[16x4] holds lane partials (lanes 0-15 -> K=0 slot, lanes 16-31 -> K=2 slot,
// per the 32-bit A-matrix layout); B = ones(4x16) => D[m,n] = rowsum(m) =
// partial(lane m) + partial(lane m+16). Each lane folds its 8 D VGPRs
// (8 rows of its half) and one shfl_xor(16) merges the halves.
__device__ __forceinline__ float wave_sum32(float x) {
  v2f a; a[0] = x;   a[1] = 0.0f;
  v2f ones; ones[0] = 1.0f; ones[1] = 1.0f;
  v8f c = {};
  c = __builtin_amdgcn_wmma_f32_16x16x4_f32(false, a, false, ones,
                                            (short)0, c, false, false);
  float s = c[0] + c[1] + c[2] + c[3] + c[4] + c[5] + c[6] + c[7];
  s += __shfl_xor(s, 16, 32);
  return s;
}

// Block-wide (32 waves) sum with broadcast; WMMA wave reduce -> LDS -> WMMA.
__device__ __forceinline__ float block_sum(float x, float* red) {
  const int lane = threadIdx.x & 31;
  const int wv   = threadIdx.x >> 5;
  float w = wave_sum32(x);
  if (lane == 0) red[wv] = w;
  __syncthreads();
  if (wv == 0) {
    float t = wave_sum32(red[lane]);
    if (lane == 0) red[NWAVES] = t;
  }
  __syncthreads();
  float r = red[NWAVES];
  __syncthreads();
  return r;
}

// Monotone float<->uint order-preserving mapping (handles -inf for positives).
__device__ __forceinline__ unsigned key_of(float f) {
  unsigned b = __float_as_uint(f);
  return (b & 0x80000000u) ? ~b : (b | 0x80000000u);
}
__device__ __forceinline__ float float_of(unsigned k) {
  unsigned b = (k & 0x80000000u) ? (k ^ 0x80000000u) : ~k;
  return __uint_as_float(b);
}

__global__ __launch_bounds__(BLOCK)
void multibox_row_kernel(const float* __restrict__ conf,
                         const float* __restrict__ pred,
                         const int*   __restrict__ labels,
                         const float* __restrict__ gt,
                         const float* __restrict__ mask,
                         float* __restrict__ part /* [BATCH][4] */) {
  __shared__ unsigned s_key[NPRIOR];      // ~34 KB of the 320 KB WGP LDS
  __shared__ float    s_red[NWAVES + 1];

  const int    b   = blockIdx.x;
  const int    tid = threadIdx.x;
  const size_t row = (size_t)b * NPRIOR;

  float reg_acc = 0.f, cpos_acc = 0.f, msum_acc = 0.f, pos_acc = 0.f;

  // ---- Pass 1: stream everything once (NT hints: data is single-use) ----
  for (int p = tid; p < NPRIOR; p += BLOCK) {
    const float* cp = conf + (row + p) * NCLS;
    if (p + BLOCK < NPRIOR) __builtin_prefetch(cp + (size_t)BLOCK * NCLS, 0, 1);

    const int lbl = labels[row + p];
    float v[NCLS];
    float m = -INFINITY, x0 = 0.f, xl = 0.f;
#pragma unroll
    for (int c = 0; c < NCLS; ++c) {
      float x = __builtin_nontemporal_load(cp + c);
      v[c] = x;
      m = fmaxf(m, x);
      if (c == 0) x0 = x;
      xl = (c == lbl) ? x : xl;   // select avoids dynamic register indexing
    }
    float s = 0.f;
#pragma unroll
    for (int c = 0; c < NCLS; ++c) s += __expf(v[c] - m);
    const float lse = m + __logf(s);
    const float bg  = lse - x0;                 // -log_softmax class 0

    const bool  pos = lbl > 0;
    const float pf  = pos ? 1.f : 0.f;
    const float w   = mask[row + p];
    msum_acc += w;
    pos_acc  += pf;
    cpos_acc += w * pf * (lse - xl);            // per-label NLL at positives

    const float* pp = pred + (row + p) * 4;
    const float* gp = gt   + (row + p) * 4;
    float rl = 0.f;
#pragma unroll
    for (int j = 0; j < 4; ++j) {
      float d  = __builtin_nontemporal_load(pp + j) -
                 __builtin_nontemporal_load(gp + j);
      float ad = fabsf(d);
      rl += (ad < 1.f) ? 0.5f * d * d : (ad - 0.5f);
    }
    reg_acc += rl * w * pf;

    // positives excluded from negative mining via -inf
    s_key[p] = key_of(pos ? -INFINITY : bg);
  }
  __syncthreads();

  // ---- K = 3 * num_pos (exact: counts < 2^24 in f32) ----
  const float posc = block_sum(pos_acc, s_red);
  int K = NEG_RATIO * (int)(posc + 0.5f);
  if (K > NPRIOR) K = NPRIOR;

  // ---- Radix-select the K-th largest masked bg_loss, sum the top-K ----
  float neg_sum = 0.f;
  if (K > 0) {                                   // block-uniform branch
    unsigned T = 0u;
    for (int bit = 31; bit >= 0; --bit) {
      const unsigned cand = T | (1u << bit);
      float c = 0.f;
      for (int p = tid; p < NPRIOR; p += BLOCK)
        c += (s_key[p] >= cand) ? 1.f : 0.f;
      const float tot = block_sum(c, s_red);
      if ((int)(tot + 0.5f) >= K) T = cand;      // T ends as K-th largest key
    }
    float sgt = 0.f, cgt = 0.f;
    for (int p = tid; p < NPRIOR; p += BLOCK) {
      const unsigned k = s_key[p];
      if (k > T) { sgt += float_of(k); cgt += 1.f; }
    }
    const float sum_gt = block_sum(sgt, s_red);
    const float cnt_gt = block_sum(cgt, s_red);
    neg_sum = sum_gt + ((float)K - cnt_gt) * float_of(T);  // value-exact ties
  }

  // ---- Per-row partials: one slot per block, no atomics (deterministic) ----
  const float reg_tot  = block_sum(reg_acc,  s_red);
  const float cpos_tot = block_sum(cpos_acc, s_red);
  const float msum_tot = block_sum(msum_acc, s_red);
  if (tid == 0) {
    part[b * 4 + 0] = reg_tot;
    part[b * 4 + 1] = cpos_tot + neg_sum;
    part[b * 4 + 2] = msum_tot;
    part[b * 4 + 3] = 0.f;
  }
}

__global__ void multibox_finalize_kernel(const float* __restrict__ part,
                                         float* __restrict__ out) {
  const int lane = threadIdx.x;   // launched with one wave32
  float r = 0.f, cf = 0.f, ms = 0.f;
  for (int b = lane; b < BATCH; b += 32) {
    r  += part[b * 4 + 0];
    cf += part[b * 4 + 1];
    ms += part[b * 4 + 2];
  }
  r  = wave_sum32(r);
  cf = wave_sum32(cf);
  ms = wave_sum32(ms);
  if (lane == 0) {
    out[0] = r  / ms;   // regression_loss / all_batch_pos
    out[1] = cf / ms;   // confidence_loss / all_batch_pos
  }
}

extern "C" void kernel_launch(void* const* d_in, const int* in_sizes, int n_in,
                              void* d_out, int out_size, void* d_ws, size_t ws_size,
                              hipStream_t stream) {
  const float* conf   = (const float*)d_in[0];
  const float* pred   = (const float*)d_in[1];
  const int*   labels = (const int*)  d_in[2];
  const float* gt     = (const float*)d_in[3];
  const float* mask   = (const float*)d_in[4];
  float* part = (float*)d_ws;          // BATCH*4 floats, fully rewritten each call
  float* out  = (float*)d_out;

  multibox_row_kernel<<<BATCH, BLOCK, 0, stream>>>(conf, pred, labels, gt, mask, part);
  multibox_finalize_kernel<<<1, 32, 0, stream>>>(part, out);

  (void)in_sizes; (void)n_in; (void)out_size; (void)ws_size;
}